// Encoder_63797444215173
// MI455X (gfx1250) — compile-verified
//
#include <hip/hip_runtime.h>
#include <hip/hip_bf16.h>
#include <math.h>

// ---------------------------------------------------------------------------
// 2-layer GCN for MI455X (gfx1250, wave32).
//   inputs: x[N,128] f32, W1[128,128], b1[128], W2[64,128], b2[64],
//           edge_index[2,E] int
//   out:    [N,64] f32
// Dense transforms: native fp32 WMMA (V_WMMA_F32_16X16X4_F32), weights staged
// in LDS (padded, bank-conflict-free) so the WMMA chain waits on short
// ds-loads instead of global loads.
// Aggregation: one wave per edge + hardware f32 atomics (L2-resident).
// ---------------------------------------------------------------------------

typedef float v2f __attribute__((ext_vector_type(2)));
typedef float v8f __attribute__((ext_vector_type(8)));

// ---------------- degree / norm ----------------

__global__ __launch_bounds__(256) void k_deg_init(float* __restrict__ deg, int N) {
    int i = blockIdx.x * 256 + threadIdx.x;
    if (i < N) deg[i] = 1.0f;            // self-loop contributes 1
}

__global__ __launch_bounds__(256) void k_deg_edges(const int* __restrict__ dst,
                                                   float* __restrict__ deg, int E) {
    int e = blockIdx.x * 256 + threadIdx.x;
    if (e < E) unsafeAtomicAdd(&deg[dst[e]], 1.0f);
}

__global__ __launch_bounds__(256) void k_rsqrt(float* __restrict__ d, int N) {
    int i = blockIdx.x * 256 + threadIdx.x;
    if (i < N) d[i] = rsqrtf(d[i]);
}

// ---------------- dense GEMM via fp32 WMMA, W staged in LDS ----------------
// C[N, NT*16] = act(A[N,KT]) @ W[NT*16, KT]^T
// Block = 8 waves; each wave owns a 16-row strip and all NT column tiles.
// LDS holds W padded to stride S = KT+4 floats: S % 64 == 4, so a wave's
// ds_load_b64 B-fetch (16 rows x 2 dwords x 2 halves) touches 64 distinct
// banks -> conflict-free.
// A-frag (ISA 7.12.2, fp32 A 16x4): lanes 0-15 hold K=k0,k0+1 for row M=lane;
// lanes 16-31 hold K=k0+2,k0+3.  C/D: VGPR g -> row g+8*half, col lane&15.

template <int NT, int KT, bool RELU>
__global__ __launch_bounds__(256) void k_gemm_wmma(const float* __restrict__ A,
                                                   const float* __restrict__ Wg,
                                                   float* __restrict__ C,
                                                   int N) {
    constexpr int Fout = NT * 16;
    constexpr int S    = KT + 4;         // padded LDS row stride (floats)
    extern __shared__ float smem[];      // Fout * S floats (dynamic LDS)

    // cooperative fill of W into padded LDS (before any divergent exit)
    for (int idx = threadIdx.x; idx < Fout * KT; idx += 256) {
        int r = idx / KT;                // KT is a power of two -> shifts
        int c = idx & (KT - 1);
        smem[r * S + c] = Wg[idx];
    }
    __syncthreads();

    int wave = (blockIdx.x * blockDim.x + threadIdx.x) >> 5;
    int lane = threadIdx.x & 31;
    int half = lane >> 4;                // 0: K pair (k0,k0+1), 1: (k0+2,k0+3)
    int l16  = lane & 15;
    int row0 = wave * 16;
    if (row0 < N) {
        int arow = row0 + l16;
        if (arow >= N) arow = N - 1;     // clamp (harmless duplicate)
        const float* Ap = A + (size_t)arow * KT + 2 * half;
        const float* Bp = smem + (size_t)l16 * S + 2 * half;

        v8f acc[NT];
#pragma unroll
        for (int j = 0; j < NT; ++j) acc[j] = {};

        for (int k0 = 0; k0 < KT; k0 += 4) {
            v2f a = *(const v2f*)(Ap + k0);
            if (RELU) { a[0] = fmaxf(a[0], 0.0f); a[1] = fmaxf(a[1], 0.0f); }
            // batch all B-fragment LDS loads, then run the WMMA chain
            v2f b[NT];
#pragma unroll
            for (int j = 0; j < NT; ++j)
                b[j] = *(const v2f*)(Bp + (size_t)j * 16 * S + k0);
#pragma unroll
            for (int j = 0; j < NT; ++j)
                acc[j] = __builtin_amdgcn_wmma_f32_16x16x4_f32(
                    /*neg_a=*/false, a, /*neg_b=*/false, b[j],
                    /*c_mod=*/(short)0, acc[j],
                    /*reuse_a=*/false, /*reuse_b=*/false);
        }

#pragma unroll
        for (int j = 0; j < NT; ++j) {
#pragma unroll
            for (int g = 0; g < 8; ++g) {
                int row = row0 + g + 8 * half;
                if (row < N) C[(size_t)row * Fout + j * 16 + l16] = acc[j][g];
            }
        }
    }
}

// ---------------- aggregation ----------------
// out[n,f] = bias[f] + dinv[n]^2 * H[n,f]      (self-loop term, plain store)

template <int F>
__global__ __launch_bounds__(256) void k_agg_init(const float* __restrict__ H,
                                                  const float* __restrict__ dinv,
                                                  const float* __restrict__ bias,
                                                  float* __restrict__ out, int N) {
    size_t i = (size_t)blockIdx.x * 256 + threadIdx.x;
    if (i >= (size_t)N * F) return;
    int n = (int)(i / F);
    int f = (int)(i % F);
    float di = dinv[n];
    out[i] = bias[f] + di * di * H[i];
}

// out[dst,:] += dinv[src]*dinv[dst] * H[src,:]   — one wave per edge,
// lane handles F/32 contiguous floats; hardware global_atomic_add_f32.

template <int F>
__global__ __launch_bounds__(256) void k_agg_edges(const float* __restrict__ H,
                                                   const float* __restrict__ dinv,
                                                   const int* __restrict__ src,
                                                   const int* __restrict__ dst,
                                                   float* __restrict__ out, int E) {
    int e    = (blockIdx.x * blockDim.x + threadIdx.x) >> 5;
    int lane = threadIdx.x & 31;
    if (e >= E) return;
    int s = src[e];
    int d = dst[e];
    float coef = dinv[s] * dinv[d];

    constexpr int V = F / 32;                    // 4 (F=128) or 2 (F=64)
    const float* hp = H   + (size_t)s * F + lane * V;
    float*       op = out + (size_t)d * F + lane * V;
    if constexpr (V == 4) {
        float4 v = *(const float4*)hp;
        unsafeAtomicAdd(op + 0, coef * v.x);
        unsafeAtomicAdd(op + 1, coef * v.y);
        unsafeAtomicAdd(op + 2, coef * v.z);
        unsafeAtomicAdd(op + 3, coef * v.w);
    } else {
        float2 v = *(const float2*)hp;
        unsafeAtomicAdd(op + 0, coef * v.x);
        unsafeAtomicAdd(op + 1, coef * v.y);
    }
}

// ---------------- launch ----------------

extern "C" void kernel_launch(void* const* d_in, const int* in_sizes, int n_in,
                              void* d_out, int out_size, void* d_ws, size_t ws_size,
                              hipStream_t stream) {
    const float* x  = (const float*)d_in[0];
    const float* W1 = (const float*)d_in[1];
    const float* b1 = (const float*)d_in[2];
    const float* W2 = (const float*)d_in[3];
    const float* b2 = (const float*)d_in[4];
    const int*   ei = (const int*)d_in[5];       // [2,E]: row0=src, row1=dst

    const int IN = 128, HID = 128, OUT = 64;
    int N = in_sizes[0] / IN;
    int E = in_sizes[5] / 2;
    const int* src = ei;
    const int* dst = ei + E;

    // workspace: dinv[N] | bufH[N*128] | bufA[N*128]
    float* dinv = (float*)d_ws;
    size_t off1 = ((size_t)N * sizeof(float) + 255) & ~(size_t)255;
    float* bufH = (float*)((char*)d_ws + off1);
    float* bufA = bufH + (size_t)N * HID;
    float* out  = (float*)d_out;

    unsigned bN   = (unsigned)((N + 255) / 256);
    unsigned bE   = (unsigned)((E + 255) / 256);
    unsigned bEw  = (unsigned)((E + 7) / 8);                 // 8 waves/block
    unsigned bGem = (unsigned)(((N + 15) / 16 + 7) / 8);     // 8 waves/block

    size_t lds1 = (size_t)(8 * 16) * (128 + 4) * sizeof(float);  // 67,584 B
    size_t lds2 = (size_t)(4 * 16) * (128 + 4) * sizeof(float);  // 33,792 B

    // degrees -> dinv
    k_deg_init <<<bN,  256, 0, stream>>>(dinv, N);
    k_deg_edges<<<bE,  256, 0, stream>>>(dst, dinv, E);
    k_rsqrt    <<<bN,  256, 0, stream>>>(dinv, N);

    // layer 1: H1 = x @ W1^T ; A1 = b1 + D^-1/2 A D^-1/2 H1
    k_gemm_wmma<8, 128, false><<<bGem, 256, lds1, stream>>>(x, W1, bufH, N);
    k_agg_init<128><<<(unsigned)(((size_t)N * HID + 255) / 256), 256, 0, stream>>>(
        bufH, dinv, b1, bufA, N);
    k_agg_edges<128><<<bEw, 256, 0, stream>>>(bufH, dinv, src, dst, bufA, E);

    // layer 2: H2 = relu(A1) @ W2^T (ReLU fused into GEMM load); aggregate at 64
    k_gemm_wmma<4, 128, true><<<bGem, 256, lds2, stream>>>(bufA, W2, bufH, N);
    k_agg_init<64><<<(unsigned)(((size_t)N * OUT + 255) / 256), 256, 0, stream>>>(
        bufH, dinv, b2, out, N);
    k_agg_edges<64><<<bEw, 256, 0, stream>>>(bufH, dinv, src, dst, out, E);
}